// NewNormalizationUnit_23407571764169
// MI455X (gfx1250) — compile-verified
//
#include <hip/hip_runtime.h>
#include <stdint.h>

// -----------------------------------------------------------------------------
// SNN "NewNormalizationUnit": boolean circuit over 4M x 8 spike bits.
// Memory-bound (416 MB @ 23.3 TB/s ~= 18us floor). WMMA not applicable
// (elementwise boolean, no contraction). CDNA5 features used instead:
//   * GLOBAL_LOAD_ASYNC_TO_LDS_B128 double-buffered input streaming (ASYNCcnt)
//   * s_wait_asynccnt pacing
//   * non-temporal (NT) B128 loads/stores for the streamed regions
// -----------------------------------------------------------------------------

typedef unsigned int u32;
typedef u32   u32x4 __attribute__((ext_vector_type(4)));
typedef float f32x4 __attribute__((ext_vector_type(4)));
typedef int   v4i   __attribute__((vector_size(16)));   // matches builtin param

#define AS1 __attribute__((address_space(1)))
#define AS3 __attribute__((address_space(3)))

#if defined(__has_builtin)
#if __has_builtin(__builtin_amdgcn_global_load_async_to_lds_b128) && \
    __has_builtin(__builtin_amdgcn_s_wait_asynccnt)
#define USE_ASYNC_LDS 1
#endif
#endif

__device__ __forceinline__ float b2f(u32 bits, int i) {
  return ((bits >> i) & 1u) ? 1.0f : 0.0f;
}

// Full circuit for one row, on packed bits (inputs are exactly 0.0/1.0 so all
// heaviside gates reduce to boolean logic).
//   h = index of MSB; shift = 7-h (0 if pb==0)
//   exp LUT per h: {26,27,28,29,30,31,0,1}; pb==0 -> 25   (verified vs e0..e4)
__device__ __forceinline__ void row_logic(u32 pb, u32& pn, u32& e, u32& sh,
                                          u32& st, u32& ov) {
  u32 h = 31u - (u32)__clz((int)(pb | 1u));      // pb==0 -> h=0 (overridden)
  sh = pb ? (7u - h) : 0u;
  e  = pb ? (u32)((0x01001F1E1D1C1B1AULL >> (h * 8u)) & 0xFFull) : 25u;
  pn = ((pb >> 1) << sh) & 0xFFu;                // barrel-shift of p_pre
  st = pb & 1u;                                  // sticky_extra = p0
  ov = (pb >> 7) & 1u;                           // overflow     = p7
}

// Compute + store 4 consecutive rows (row0 % 4 == 0, B % 4 == 0) so every
// output region is whole, 16B-aligned float4s. All stores non-temporal.
__device__ __forceinline__ void compute_store4(const u32x4* va, const u32x4* vb,
                                               float* out, size_t B,
                                               size_t row0) {
  f32x4 pnv[8], exv[5], shv[3];
  f32x4 stv = {}, ovv = {};
  float* pnf = (float*)pnv;
  float* exf = (float*)exv;
  float* shf = (float*)shv;
#pragma unroll
  for (int r = 0; r < 4; ++r) {
    u32 pb = (va[r].x ? 1u : 0u)  | (va[r].y ? 2u : 0u)  |
             (va[r].z ? 4u : 0u)  | (va[r].w ? 8u : 0u)  |
             (vb[r].x ? 16u : 0u) | (vb[r].y ? 32u : 0u) |
             (vb[r].z ? 64u : 0u) | (vb[r].w ? 128u : 0u);
    u32 pn, e, sh, st, ov;
    row_logic(pb, pn, e, sh, st, ov);
#pragma unroll
    for (int i = 0; i < 8; ++i) pnf[r * 8 + i] = b2f(pn, i);
#pragma unroll
    for (int i = 0; i < 5; ++i) exf[r * 5 + i] = b2f(e, i);
#pragma unroll
    for (int i = 0; i < 3; ++i) shf[r * 3 + i] = b2f(sh, i);
    stv[r] = (float)st;
    ovv[r] = (float)ov;
  }
  // Output layout: [B,8] p_norm | [B,5] exp_adj | [B] sticky | [B] ovf | [B,3] shift
  f32x4* pnp = (f32x4*)(out + row0 * 8);
  f32x4* exq = (f32x4*)(out + B * 8 + row0 * 5);
  f32x4* stp = (f32x4*)(out + B * 13 + row0);
  f32x4* ovp = (f32x4*)(out + B * 14 + row0);
  f32x4* shp = (f32x4*)(out + B * 15 + row0 * 3);
#pragma unroll
  for (int k = 0; k < 8; ++k) __builtin_nontemporal_store(pnv[k], pnp + k);
#pragma unroll
  for (int k = 0; k < 5; ++k) __builtin_nontemporal_store(exv[k], exq + k);
  __builtin_nontemporal_store(stv, stp);
  __builtin_nontemporal_store(ovv, ovp);
#pragma unroll
  for (int k = 0; k < 3; ++k) __builtin_nontemporal_store(shv[k], shp + k);
}

// Scalar fallback for ragged tails / non-multiple-of-4 B.
__device__ __forceinline__ void scalar_row(const u32* P, float* out, size_t B,
                                           size_t r) {
  u32 pb = 0;
  for (int i = 0; i < 8; ++i) pb |= (P[r * 8 + i] != 0u ? 1u : 0u) << i;
  u32 pn, e, sh, st, ov;
  row_logic(pb, pn, e, sh, st, ov);
  for (int i = 0; i < 8; ++i) out[r * 8 + i] = b2f(pn, i);
  for (int i = 0; i < 5; ++i) out[B * 8 + r * 5 + i] = b2f(e, i);
  out[B * 13 + r] = (float)st;
  out[B * 14 + r] = (float)ov;
  for (int i = 0; i < 3; ++i) out[B * 15 + r * 3 + i] = b2f(sh, i);
}

#if defined(USE_ASYNC_LDS)
// ---- Async-to-LDS double-buffered streaming path (CDNA5) --------------------
// Tile = 1024 rows = 256 threads * 4 rows = 32KB input. LDS layout [k][tid]
// (16B/lane stride) makes ds_load_b128 bank-conflict free. Each lane reads
// only LDS its own async loads wrote -> no barriers, just ASYNCcnt pacing.
__device__ __forceinline__ void issue_tile_async(const u32* P, int t, int tid,
                                                 u32x4 (*buf)[256]) {
  const u32x4* src = (const u32x4*)P + (size_t)t * 2048 + (size_t)tid * 8;
#pragma unroll
  for (int k = 0; k < 8; ++k) {
    // flat->AS1 is bit-identical; flat->AS3 is the low-32-bit LDS offset
    __builtin_amdgcn_global_load_async_to_lds_b128(
        (AS1 v4i*)(uintptr_t)(src + k),
        (AS3 v4i*)(u32)(uintptr_t)(&buf[k][tid]),
        0, 0);
  }
}

__global__ __launch_bounds__(256) void snn_norm_pipe(const u32* __restrict__ P,
                                                     float* __restrict__ out,
                                                     size_t B, int numTiles) {
  __shared__ u32x4 lds[2][8][256];  // 64KB double buffer (of 320KB/WGP)
  const int tid = (int)threadIdx.x;
  const int stride = (int)gridDim.x;
  int t = (int)blockIdx.x;
  if (t < numTiles) issue_tile_async(P, t, tid, lds[0]);
  int g = 0;
  for (; t < numTiles; t += stride, ++g) {
    const int buf = g & 1;
    const int tn = t + stride;
    if (tn < numTiles) {
      asm volatile("" ::: "memory");  // keep prior LDS reads before re-issue
      issue_tile_async(P, tn, tid, lds[buf ^ 1]);
      __builtin_amdgcn_s_wait_asynccnt(8);  // async loads retire in order:
                                            // <=8 left => current tile landed
    } else {
      __builtin_amdgcn_s_wait_asynccnt(0);
    }
    u32x4 va[4], vb[4];
#pragma unroll
    for (int r = 0; r < 4; ++r) {
      va[r] = lds[buf][2 * r][tid];      // ds_load_b128, conflict-free
      vb[r] = lds[buf][2 * r + 1][tid];
    }
    const size_t row0 = (size_t)t * 1024 + (size_t)tid * 4;
    compute_store4(va, vb, out, B, row0);
  }
}
#endif  // USE_ASYNC_LDS

// ---- Direct-load fallback (also handles ragged shapes) ----------------------
__global__ __launch_bounds__(256) void snn_norm_simple(const u32* __restrict__ P,
                                                       float* __restrict__ out,
                                                       size_t B) {
  const size_t tg = (size_t)blockIdx.x * 256 + threadIdx.x;
  const size_t row0 = tg * 4;
  if (((B & 3) == 0) && row0 + 4 <= B) {
    const u32x4* gp = (const u32x4*)P + row0 * 2;
    u32x4 va[4], vb[4];
#pragma unroll
    for (int r = 0; r < 4; ++r) {
      va[r] = __builtin_nontemporal_load(gp + 2 * r);
      vb[r] = __builtin_nontemporal_load(gp + 2 * r + 1);
    }
    compute_store4(va, vb, out, B, row0);
  } else {
    for (size_t r = row0; r < B && r < row0 + 4; ++r) scalar_row(P, out, B, r);
  }
}

extern "C" void kernel_launch(void* const* d_in, const int* in_sizes, int n_in,
                              void* d_out, int out_size, void* d_ws,
                              size_t ws_size, hipStream_t stream) {
  (void)n_in; (void)out_size; (void)d_ws; (void)ws_size;
  const u32* P = (const u32*)d_in[0];
  float* out = (float*)d_out;
  const size_t B = (size_t)in_sizes[0] / 8;
  if (B == 0) return;
#if defined(USE_ASYNC_LDS)
  if ((B % 1024) == 0) {
    const int numTiles = (int)(B / 1024);                 // 4096 @ B=4M
    int grid = (numTiles + 3) / 4;                        // 4 tiles/block ->
    if (grid < 1) grid = 1;                               // real double-buffer
    snn_norm_pipe<<<grid, 256, 0, stream>>>(P, out, B, numTiles);
    return;
  }
#endif
  const size_t nthreads = (B + 3) / 4;
  int blocks = (int)((nthreads + 255) / 256);
  if (blocks < 1) blocks = 1;
  snn_norm_simple<<<blocks, 256, 0, stream>>>(P, out, B);
}